// ContentAdjMasked_29368986370083
// MI455X (gfx1250) — compile-verified
//
#include <hip/hip_runtime.h>
#include <hip/hip_bf16.h>

typedef float v2f __attribute__((ext_vector_type(2)));
typedef float v4f __attribute__((ext_vector_type(4)));
typedef float v8f __attribute__((ext_vector_type(8)));
typedef unsigned int u32x4 __attribute__((ext_vector_type(4)));
typedef int          i32x4 __attribute__((ext_vector_type(4)));
typedef int          i32x8 __attribute__((ext_vector_type(8)));

#define DIM 128
#define KNN 32
#define TAUI 5.0f     /* 1/TAU, TAU=0.2 */
#define EPS 1e-8f

#define LDS_ROW   132                      /* 128 + 4 DWORD pad -> conflict-free b64 reads */
#define WOFF_F    (128 * LDS_ROW)          /* floats per staged weight matrix */
#define SMEM_BYTES (2 * WOFF_F * 4)        /* 132 KB dynamic LDS */

#define HAVE_TDM (__has_builtin(__builtin_amdgcn_tensor_load_to_lds) && \
                  __has_builtin(__builtin_amdgcn_s_wait_tensorcnt))

#if HAVE_TDM
// Issue one TDM descriptor: 2D tensor 128x128 f32 at gsrc, tile = whole tensor,
// destination LDS at byte offset lds_off, padding 4 DWORDs after every 128 DWORDs
// (row stride 132 floats in LDS).
__device__ __forceinline__ void tdm_load_weight(const float* gsrc, unsigned lds_off)
{
    unsigned long long ga = (unsigned long long)(size_t)gsrc;
    u32x4 g0;
    g0[0] = 1u;                                            // count=1, user mode, no gather
    g0[1] = lds_off;                                       // lds_addr (bytes)
    g0[2] = (unsigned)(ga & 0xFFFFFFFFu);                  // global_addr[31:0]
    g0[3] = (unsigned)((ga >> 32) & 0x01FFFFFFu)           // global_addr[56:32]
          | 0x80000000u;                                   // type = 2 ("image")
    i32x8 g1;
    g1[0] = (2 << 16)        /* data_size = 4 B */
          | (1 << 20)        /* pad_enable */
          | (6 << 22)        /* pad_interval: 128 DWORDs */
          | (3 << 25);       /* pad_amount : 4 DWORDs   */
    g1[1] = (int)(128u << 16);   // tensor_dim0 = 128   (bits 79:48)
    g1[2] = (int)(128u << 16);   // tensor_dim1 = 128   (bits 111:80)
    g1[3] = (int)(128u << 16);   // tile_dim0   = 128   (bits 127:112)
    g1[4] = 128;                 // tile_dim1   = 128   (bits 143:128), tile_dim2 = 0
    g1[5] = 128;                 // tensor_dim0_stride = 128 (bits 207:160)
    g1[6] = 0;
    g1[7] = 0;                   // tensor_dim1_stride unused (2D tile)
    i32x4 z4 = {0, 0, 0, 0};
#if defined(__clang_major__) && __clang_major__ >= 23
    i32x8 z8 = {0, 0, 0, 0, 0, 0, 0, 0};
    __builtin_amdgcn_tensor_load_to_lds(g0, g1, z4, z4, z8, 0);
#else
    __builtin_amdgcn_tensor_load_to_lds(g0, g1, z4, z4, 0);
#endif
}
#endif

// ---------------------------------------------------------------------------
// Kernel 1: fused projection  Q = H @ Wq^T,  K = H @ Wk^T  via fp32 WMMA.
// Wq/Wk staged in LDS once per block (TDM tensor_load_to_lds when available),
// 8 waves/block, each wave owns a 16-row strip across all 128 columns.
// ---------------------------------------------------------------------------
__global__ __launch_bounds__(256)
void proj_qk_wmma(const float* __restrict__ H,
                  const float* __restrict__ Wq,
                  const float* __restrict__ Wk,
                  float* __restrict__ Q,
                  float* __restrict__ Km,
                  int L)
{
    extern __shared__ float smem[];        // [2][128][LDS_ROW]

    const int lane = threadIdx.x & 31;
    const int wave = threadIdx.x >> 5;
    const int m0   = blockIdx.x * 128 + wave * 16;

    // ---- stage both weight matrices into LDS ----
#if HAVE_TDM
    if (threadIdx.x == 0) {
        const unsigned base = (unsigned)(size_t)(void*)smem;   // LDS offset = addr[31:0]
        tdm_load_weight(Wq, base);
        tdm_load_weight(Wk, base + (unsigned)(WOFF_F * 4));
        __builtin_amdgcn_s_wait_tensorcnt(0);
    }
#else
    for (int c = threadIdx.x; c < 2 * 128 * 32; c += 256) {    // 16B chunks
        const int m  = c >> 12;            // matrix select
        const int r  = (c >> 5) & 127;     // row
        const int ch = c & 31;             // 16B chunk in row
        const float* src = (m ? Wk : Wq) + (size_t)r * DIM + ch * 4;
        v4f t = *(const v4f*)src;
        *(v4f*)(smem + (size_t)m * WOFF_F + (size_t)r * LDS_ROW + ch * 4) = t;
    }
#endif
    __syncthreads();

    const int c0 = (lane < 16) ? 0 : 2;            // K sub-offset for this half
    const int rA = min(m0 + (lane & 15), L - 1);   // clamp tail rows (stores predicated)
    const float* hrow = H + (size_t)rA * DIM;

    // Hoist all 32 A-fragments for the full K=128 loop (64 VGPRs).
    v2f a[32];
#pragma unroll
    for (int kk = 0; kk < 32; ++kk)
        a[kk] = *(const v2f*)(hrow + kk * 4 + c0);

    const int nlane = lane & 15;
    const int rbase = m0 + ((lane >> 4) << 3);     // store row base per half-wave

#pragma unroll 1
    for (int nt = 0; nt < 8; ++nt) {
        const int n = nt * 16 + nlane;
        const float* wqr = smem + (size_t)n * LDS_ROW + c0;          // B[j][n] = W[n][j]
        const float* wkr = smem + WOFF_F + (size_t)n * LDS_ROW + c0;

        v8f cq = {};
        v8f ck = {};
#pragma unroll
        for (int kk = 0; kk < 32; ++kk) {
            v2f bq = *(const v2f*)(wqr + kk * 4);   // ds_load_b64, conflict-free
            v2f bk = *(const v2f*)(wkr + kk * 4);
            cq = __builtin_amdgcn_wmma_f32_16x16x4_f32(false, a[kk], false, bq,
                                                       (short)0, cq, false, false);
            ck = __builtin_amdgcn_wmma_f32_16x16x4_f32(false, a[kk], false, bk,
                                                       (short)0, ck, false, false);
        }

#pragma unroll
        for (int i = 0; i < 8; ++i) {
            const int r = rbase + i;
            if (r < L) {
                Q [(size_t)r * DIM + n] = cq[i];
                Km[(size_t)r * DIM + n] = ck[i];
            }
        }
    }
}

// ---------------------------------------------------------------------------
// Kernel 2: one wave32 per row l. Lane k owns neighbor k (KNN == 32 lanes).
// Each lane streams K[idx[l,k]] (L2-resident) in b128 chunks and accumulates
// the 128-dim dot in registers; softmax / mix / renorm use wave reductions.
// ---------------------------------------------------------------------------
__global__ __launch_bounds__(256)
void gather_softmax_mix(const float* __restrict__ Q,
                        const float* __restrict__ Km,
                        const float* __restrict__ mix,
                        const float* __restrict__ knn_w,
                        const long long* __restrict__ knn_idx,
                        float* __restrict__ vals,
                        long long* __restrict__ rows,
                        long long* __restrict__ cols,
                        int L)
{
    const int lane = threadIdx.x & 31;
    const int wave = threadIdx.x >> 5;
    const int l    = blockIdx.x * 8 + wave;
    if (l >= L) return;

    const long long nb64 = knn_idx[(size_t)l * KNN + lane];
    const v4f* kr = (const v4f*)(Km + (size_t)nb64 * DIM);
    const v4f* qr = (const v4f*)(Q  + (size_t)l    * DIM);
    __builtin_prefetch(kr, 0, 3);   // global_prefetch_b8 for the gathered row

    float sim = 0.f;
#pragma unroll
    for (int j = 0; j < DIM / 4; ++j) {
        v4f kv = kr[j];
        v4f qv = qr[j];               // uniform across lanes -> broadcast from cache
        sim += kv.x * qv.x + kv.y * qv.y + kv.z * qv.z + kv.w * qv.w;
    }

    // softmax(sim / TAU) over the 32 lanes
    float x = sim * TAUI;
    float m = x;
#pragma unroll
    for (int off = 16; off >= 1; off >>= 1)
        m = fmaxf(m, __shfl_xor(m, off, 32));
    float e = __expf(x - m);
    float s = e;
#pragma unroll
    for (int off = 16; off >= 1; off >>= 1)
        s += __shfl_xor(s, off, 32);
    const float A = e / s;

    const float beta = 1.0f / (1.0f + __expf(-mix[0]));
    float w = (1.0f - beta) * knn_w[(size_t)l * KNN + lane] + beta * A;
    float ws = w;
#pragma unroll
    for (int off = 16; off >= 1; off >>= 1)
        ws += __shfl_xor(ws, off, 32);
    w = w / (ws + EPS);

    const size_t o = (size_t)l * KNN + lane;
    vals[o] = w;
    rows[o] = (long long)l;
    cols[o] = nb64;
}

// ---------------------------------------------------------------------------
extern "C" void kernel_launch(void* const* d_in, const int* in_sizes, int n_in,
                              void* d_out, int out_size, void* d_ws, size_t ws_size,
                              hipStream_t stream)
{
    const float*     H       = (const float*)d_in[0];
    const float*     Wq      = (const float*)d_in[1];
    const float*     Wk      = (const float*)d_in[2];
    const float*     mix     = (const float*)d_in[3];
    const float*     knn_w   = (const float*)d_in[4];
    const long long* knn_idx = (const long long*)d_in[5];

    const int    L  = in_sizes[0] / DIM;
    const size_t LK = (size_t)L * KNN;

    // scratch: Q then K, each L*DIM fp32 (25.6 MB each)
    float* Q  = (float*)d_ws;
    float* Km = Q + (size_t)L * DIM;

    // output: vals f32 [L*K] | rows i64 [L*K] | cols i64 [L*K], flat in bytes
    float*     vals = (float*)d_out;
    long long* rows = (long long*)(vals + LK);
    long long* cols = rows + LK;

    dim3 blk(256);
    dim3 g1((L + 127) / 128);
    proj_qk_wmma<<<g1, blk, SMEM_BYTES, stream>>>(H, Wq, Wk, Q, Km, L);

    dim3 g2((L + 7) / 8);
    gather_softmax_mix<<<g2, blk, 0, stream>>>(Q, Km, mix, knn_w, knn_idx,
                                               vals, rows, cols, L);
}